// GCNEncoder_50560355009131
// MI455X (gfx1250) — compile-verified
//
#include <hip/hip_runtime.h>
#include <hip/hip_bf16.h>

typedef float v2f __attribute__((ext_vector_type(2)));
typedef float v8f __attribute__((ext_vector_type(8)));

#define DF 128

// ---------------------------------------------------------------------------
// Y[M,128] = X[M,128] @ W[128,128]  via V_WMMA_F32_16X16X4_F32 (fp32 matrix op)
// One wave computes a 16-row x 128-col strip (8 WMMA tiles of 16x16).
// W is staged in LDS, interleaved so B-fragments are single 8B loads:
//   lw[(k>>1)*256 + n*2 + (k&1)] = W[k][n]
// A fragment (16x4 f32): lane m=lane&15, khalf=lane>>4:
//   a = { X[m][k0+2*khalf], X[m][k0+2*khalf+1] }
// B fragment (4x16 f32): VGPR j holds K = j + 2*khalf, col = n*16+m
// C/D: VGPR r -> row r + 8*khalf, col = n*16+m
// M must be a multiple of 16 (100000 and 50000 both are).
// ---------------------------------------------------------------------------
__global__ __launch_bounds__(256) void gemm128_wmma(const float* __restrict__ X,
                                                    const float* __restrict__ Wm,
                                                    float* __restrict__ Y,
                                                    int M) {
    __shared__ float lw[DF * DF];  // 64 KB
    const int t = threadIdx.x;
    for (int idx = t; idx < DF * DF; idx += 256) {
        const int k = idx >> 7;
        const int n = idx & 127;
        lw[(k >> 1) * 256 + n * 2 + (k & 1)] = Wm[idx];
    }
    __syncthreads();

    const int wave = t >> 5;
    const int lane = t & 31;
    const long tile = (long)blockIdx.x * 8 + wave;   // 16-row tile index
    if (tile * 16 >= (long)M) return;                // wave-uniform exit

    const int m = lane & 15;
    const int khalf = lane >> 4;                     // 0 or 1
    const float* xrow = X + tile * 16 * DF + (long)m * DF + khalf * 2;

    v8f zero = {};
    v8f acc[8];
#pragma unroll
    for (int n = 0; n < 8; ++n) acc[n] = zero;

    for (int k0 = 0; k0 < DF; k0 += 4) {
        v2f a = *(const v2f*)(xrow + k0);
        const float* lb = lw + ((k0 >> 1) + khalf) * 256;
#pragma unroll
        for (int n = 0; n < 8; ++n) {
            v2f bf = *(const v2f*)(lb + (n * 16 + m) * 2);
            acc[n] = __builtin_amdgcn_wmma_f32_16x16x4_f32(
                false, a, false, bf, (short)0, acc[n], false, false);
        }
    }

    float* yrow = Y + tile * 16 * DF;
#pragma unroll
    for (int n = 0; n < 8; ++n) {
#pragma unroll
        for (int r = 0; r < 8; ++r) {
            yrow[(long)(r + khalf * 8) * DF + n * 16 + m] = acc[n][r];
        }
    }
}

// ---------------------------------------------------------------------------
// out[r,:] = bias[:]  for r in [0,M)
// ---------------------------------------------------------------------------
__global__ void bias_init(float* __restrict__ out, const float* __restrict__ bias,
                          long total) {
    long i = (long)blockIdx.x * blockDim.x + threadIdx.x;
    if (i < total) out[i] = bias[i & (DF - 1)];
}

// ---------------------------------------------------------------------------
// Edge scatter: out[sidx[e], :] += w[e] * X[gidx[e], :]
// One wave per edge; 32 lanes x float4 cover the 128-col row; L2 fp32 atomics.
// ---------------------------------------------------------------------------
__global__ void spmm_scatter(const float* __restrict__ X,
                             const int* __restrict__ gidx,
                             const int* __restrict__ sidx,
                             const float* __restrict__ w,
                             float* __restrict__ out, int E) {
    long i = (long)blockIdx.x * blockDim.x + threadIdx.x;
    long total = (long)E * 32;
    if (i >= total) return;
    const int e = (int)(i >> 5);
    const int c = (int)(i & 31) * 4;
    const float we = w[e];
    const float4 x = *(const float4*)(X + (long)gidx[e] * DF + c);
    float* o = out + (long)sidx[e] * DF + c;
    atomicAdd(o + 0, we * x.x);
    atomicAdd(o + 1, we * x.y);
    atomicAdd(o + 2, we * x.z);
    atomicAdd(o + 3, we * x.w);
}

// cur = relu(t); sum += cur
__global__ void relu_acc(const float* __restrict__ t, float* __restrict__ cur,
                         float* __restrict__ sum, long n) {
    long i = (long)blockIdx.x * blockDim.x + threadIdx.x;
    if (i < n) {
        float r = t[i];
        r = r > 0.0f ? r : 0.0f;
        cur[i] = r;
        sum[i] += r;
    }
}

// dst1 = dst2 = src
__global__ void copy2(const float* __restrict__ src, float* __restrict__ d1,
                      float* __restrict__ d2, long n) {
    long i = (long)blockIdx.x * blockDim.x + threadIdx.x;
    if (i < n) {
        float v = src[i];
        d1[i] = v;
        d2[i] = v;
    }
}

// out = sum * scale
__global__ void finalize(const float* __restrict__ sum, float* __restrict__ out,
                         float scale, long n) {
    long i = (long)blockIdx.x * blockDim.x + threadIdx.x;
    if (i < n) out[i] = sum[i] * scale;
}

static inline int gblk(long n) { return (int)((n + 255) / 256); }

extern "C" void kernel_launch(void* const* d_in, const int* in_sizes, int n_in,
                              void* d_out, int out_size, void* d_ws, size_t ws_size,
                              hipStream_t stream) {
    const float* ufea = (const float*)d_in[0];
    const float* vfea = (const float*)d_in[1];
    const int*   eu   = (const int*)d_in[2];   // edge_user
    const int*   ei   = (const int*)d_in[3];   // edge_item
    const float* ew   = (const float*)d_in[4]; // edge_w
    const float* Wall = (const float*)d_in[5]; // [L,4,128,128]
    const float* ball = (const float*)d_in[6]; // [L,4,128]

    const int NU = in_sizes[0] / DF;
    const int NV = in_sizes[1] / DF;
    const int E  = in_sizes[2];
    const int L  = in_sizes[6] / (4 * DF);

    const long su = (long)NU * DF;
    const long sv = (long)NV * DF;

    float* ws   = (float*)d_ws;
    float* u    = ws;            // user activations (also reused for tD @ W3)
    float* usum = u + su;
    float* tC   = usum + su;     // u@W0 out, later spmm user out
    float* tD   = tC + su;       // item_ho (user-sized)
    float* v    = tD + su;
    float* vsum = v + sv;
    float* tV   = vsum + sv;     // v@W1 out, later ho_v@W2 out
    float* hoV  = tV + sv;       // user_ho (item-sized), later final item spmm out

    const int gemmBlkU = (int)((NU / 16 + 7) / 8);
    const int gemmBlkV = (int)((NV / 16 + 7) / 8);
    const long etotal  = (long)E * 32;

    // init: u=usum=ufea, v=vsum=vfea
    copy2<<<gblk(su), 256, 0, stream>>>(ufea, u, usum, su);
    copy2<<<gblk(sv), 256, 0, stream>>>(vfea, v, vsum, sv);

    for (int l = 0; l < L; ++l) {
        const float* W0 = Wall + (long)(l * 4 + 0) * DF * DF;
        const float* W1 = Wall + (long)(l * 4 + 1) * DF * DF;
        const float* W2 = Wall + (long)(l * 4 + 2) * DF * DF;
        const float* W3 = Wall + (long)(l * 4 + 3) * DF * DF;
        const float* b0 = ball + (long)(l * 4 + 0) * DF;
        const float* b1 = ball + (long)(l * 4 + 1) * DF;
        const float* b2 = ball + (long)(l * 4 + 2) * DF;
        const float* b3 = ball + (long)(l * 4 + 3) * DF;

        // 1) tC = u @ W0                       [NU,128]
        gemm128_wmma<<<gemmBlkU, 256, 0, stream>>>(u, W0, tC, NU);
        // 2) hoV = spmm_vu(tC) + b0            [NV,128]  (gather by user, scatter to item)
        bias_init<<<gblk(sv), 256, 0, stream>>>(hoV, b0, sv);
        spmm_scatter<<<gblk(etotal), 256, 0, stream>>>(tC, eu, ei, ew, hoV, E);
        // 3) tV = v @ W1                       [NV,128]
        gemm128_wmma<<<gemmBlkV, 256, 0, stream>>>(v, W1, tV, NV);
        // 4) tD = spmm_uv(tV) + b1             [NU,128]
        bias_init<<<gblk(su), 256, 0, stream>>>(tD, b1, su);
        spmm_scatter<<<gblk(etotal), 256, 0, stream>>>(tV, ei, eu, ew, tD, E);
        // 5) tV = hoV @ W2                     [NV,128]   (tV consumed in step 4)
        gemm128_wmma<<<gemmBlkV, 256, 0, stream>>>(hoV, W2, tV, NV);
        // 6) tC = spmm_uv(tV) + b2             [NU,128]   (tC consumed in step 2)
        bias_init<<<gblk(su), 256, 0, stream>>>(tC, b2, su);
        spmm_scatter<<<gblk(etotal), 256, 0, stream>>>(tV, ei, eu, ew, tC, E);
        // 7) u = tD @ W3                       [NU,128]   (u consumed in step 1)
        gemm128_wmma<<<gemmBlkU, 256, 0, stream>>>(tD, W3, u, NU);
        // 8) hoV = spmm_vu(u) + b3             [NV,128]   (hoV consumed in step 5)
        bias_init<<<gblk(sv), 256, 0, stream>>>(hoV, b3, sv);
        spmm_scatter<<<gblk(etotal), 256, 0, stream>>>(u, eu, ei, ew, hoV, E);
        // 9) u = relu(tC); usum += u           (u-as-temp consumed in step 8)
        relu_acc<<<gblk(su), 256, 0, stream>>>(tC, u, usum, su);
        // 10) v = relu(hoV); vsum += v
        relu_acc<<<gblk(sv), 256, 0, stream>>>(hoV, v, vsum, sv);
    }

    const float scale = 1.0f / (float)(L + 1);
    float* out = (float*)d_out;
    finalize<<<gblk(su), 256, 0, stream>>>(usum, out, scale, su);
    finalize<<<gblk(sv), 256, 0, stream>>>(vsum, out + su, scale, sv);
}